// MyNet_19378892440028
// MI455X (gfx1250) — compile-verified
//
#include <hip/hip_runtime.h>
#include <hip/hip_bf16.h>

#define NUM_NODES 100000
#define NUM_EDGES 400000
#define X_DIM 64
#define E_DIM 16
#define INNER 512
#define NUM_MOLS 4096
#define H2 50

typedef __attribute__((ext_vector_type(2))) float v2f;
typedef __attribute__((ext_vector_type(8))) float v8f;

// Workspace layout (floats): [0..511]=v=W2@W3, [512..591]=wc=W1@v, [592]=beta, [593]=c0
#define WS_V 0
#define WS_WC 512
#define WS_BETA 592
#define WS_C0 593

// ---------------------------------------------------------------------------
// Kernel 1: collapse weights. One block of 256 threads (8 waves).
//   v  = W2 @ W3                      [512]
//   c0 = b2 @ W3 + b3                 scalar
//   beta = b1 . v                     scalar
//   wc = W1 @ v                       [80]   <-- computed with v_wmma_f32_16x16x4_f32
// ---------------------------------------------------------------------------
__global__ void __launch_bounds__(256)
prep_kernel(const float* __restrict__ W1, const float* __restrict__ b1,
            const float* __restrict__ W2, const float* __restrict__ b2,
            const float* __restrict__ W3, const float* __restrict__ b3,
            float* __restrict__ ws) {
    __shared__ float v_s[INNER];
    __shared__ float red[256];

    const int t = threadIdx.x;

    // ---- v[j] = sum_k W2[j*50+k] * W3[k]  (2 j's per thread) ----
    float bpart = 0.f;
    for (int j = t; j < INNER; j += 256) {
        float acc = 0.f;
        #pragma unroll 10
        for (int k = 0; k < H2; ++k) acc += W2[j * H2 + k] * W3[k];
        v_s[j] = acc;
        ws[WS_V + j] = acc;
        bpart += b1[j] * acc;       // partial of beta = b1 . v
    }
    red[t] = bpart;
    __syncthreads();

    // ---- reduce beta across the block ----
    for (int off = 128; off > 0; off >>= 1) {
        if (t < off) red[t] += red[t + off];
        __syncthreads();
    }
    if (t == 0) {
        ws[WS_BETA] = red[0];
        float c0 = b3[0];
        for (int k = 0; k < H2; ++k) c0 += b2[k] * W3[k];
        ws[WS_C0] = c0;
    }
    __syncthreads();

    // ---- wc = W1 @ v via V_WMMA_F32_16X16X4_F32, wave 0 only ----
    // A tile: 16 rows of W1 x K=4.  Lane slot->K map: K = vgpr + 2*(lane>=16).
    // B tile: v[k0..k0+3] replicated across all 16 columns (same map), so
    // every column of D holds the same 16 partial dot products.
    if (t < 32) {
        const int lane = t;
        const int half = lane >> 4;          // 0: lanes 0-15, 1: lanes 16-31
        const int mrow = lane & 15;          // row within tile
        for (int tile = 0; tile < 5; ++tile) {   // 5*16 = 80 rows of W1
            v8f c = {0.f, 0.f, 0.f, 0.f, 0.f, 0.f, 0.f, 0.f};
            const int row = tile * 16 + mrow;
            const float* wrow = W1 + (long)row * INNER;
            for (int k0 = 0; k0 < INNER; k0 += 4) {
                const int kk = k0 + 2 * half;
                v2f a, b;
                a.x = wrow[kk];
                a.y = wrow[kk + 1];
                b.x = v_s[kk];
                b.y = v_s[kk + 1];
                c = __builtin_amdgcn_wmma_f32_16x16x4_f32(
                        /*neg_a=*/false, a, /*neg_b=*/false, b,
                        /*c_mod=*/(short)0, c, /*reuse_a=*/false, /*reuse_b=*/false);
            }
            // C/D layout: lane n<16 holds rows 0..7 in c[0..7]; lane n>=16 rows 8..15.
            if (lane == 0) {
                #pragma unroll
                for (int g = 0; g < 8; ++g) ws[WS_WC + tile * 16 + g] = c[g];
            } else if (lane == 16) {
                #pragma unroll
                for (int g = 0; g < 8; ++g) ws[WS_WC + tile * 16 + 8 + g] = c[g];
            }
        }
    }
}

// ---------------------------------------------------------------------------
// Kernel 2: out[m] = c0  (bias path for every molecule, incl. empty ones)
// ---------------------------------------------------------------------------
__global__ void __launch_bounds__(256)
init_out_kernel(float* __restrict__ out, const float* __restrict__ ws) {
    int i = blockIdx.x * 256 + threadIdx.x;
    if (i < NUM_MOLS) out[i] = ws[WS_C0];
}

// ---------------------------------------------------------------------------
// Kernel 3: streaming edge pass. One wave32 per edge.
//   contribution(e) = x[src[e]] . wc[0:64] + ea[e] . wc[64:80] + beta
//   atomicAdd(out[batch[dst[e]]], contribution)
// Fully coalesced: lanes 0-31 read two 128B rows of x and one 64B ea row.
// ---------------------------------------------------------------------------
__global__ void __launch_bounds__(256)
edge_kernel(const float* __restrict__ x, const int* __restrict__ edge_index,
            const float* __restrict__ ea, const int* __restrict__ batch,
            const float* __restrict__ ws, float* __restrict__ out) {
    __shared__ float wcs[80];
    __shared__ float beta_s;

    const int t = threadIdx.x;
    if (t < 80) wcs[t] = ws[WS_WC + t];
    if (t == 80) beta_s = ws[WS_BETA];
    __syncthreads();

    const int lane = t & 31;
    const int e = (blockIdx.x * 256 + t) >> 5;   // global wave id == edge id
    if (e >= NUM_EDGES) return;

    const int src = edge_index[e];               // row 0 of [2, E]
    const int dst = edge_index[NUM_EDGES + e];   // row 1
    const int mol = batch[dst];

    const float* xr = x + (long)src * X_DIM;
    float p = xr[lane] * wcs[lane] + xr[32 + lane] * wcs[32 + lane];
    if (lane < E_DIM) p += ea[(long)e * E_DIM + lane] * wcs[X_DIM + lane];

    // wave32 butterfly reduction
    #pragma unroll
    for (int off = 16; off > 0; off >>= 1) p += __shfl_xor(p, off, 32);

    if (lane == 0) atomicAdd(&out[mol], p + beta_s);
}

extern "C" void kernel_launch(void* const* d_in, const int* in_sizes, int n_in,
                              void* d_out, int out_size, void* d_ws, size_t ws_size,
                              hipStream_t stream) {
    const float* x  = (const float*)d_in[0];
    const int*   ei = (const int*)d_in[1];
    const float* ea = (const float*)d_in[2];
    const int*   bt = (const int*)d_in[3];
    const float* W1 = (const float*)d_in[4];
    const float* b1 = (const float*)d_in[5];
    const float* W2 = (const float*)d_in[6];
    const float* b2 = (const float*)d_in[7];
    const float* W3 = (const float*)d_in[8];
    const float* b3 = (const float*)d_in[9];
    float* out = (float*)d_out;
    float* ws  = (float*)d_ws;

    prep_kernel<<<1, 256, 0, stream>>>(W1, b1, W2, b2, W3, b3, ws);
    init_out_kernel<<<NUM_MOLS / 256, 256, 0, stream>>>(out, ws);
    edge_kernel<<<(NUM_EDGES * 32) / 256, 256, 0, stream>>>(x, ei, ea, bt, ws, out);
}